// Attention_test1_55748675502119
// MI455X (gfx1250) — compile-verified
//
#include <hip/hip_runtime.h>
#include <cstdint>

// ---------------------------------------------------------------------------
// MI455X (gfx1250) implementation. Memory-bound workload (~18 GFLOP vs
// ~0.55 GB traffic after fusion -> ~25us at 23.3 TB/s), so fp32 end-to-end,
// V_WMMA_F32_16X16X4_F32 for every GEMM-shaped stage, LayerNorm fused into
// GEMM B-tile loads, attention-apply folded into the projection GEMM, and
// async global->LDS copies (ASYNCcnt) for weight tiles.
// ---------------------------------------------------------------------------

#define HW   16384   // 128*128
#define WPX  128     // image width
#define CDIM 192
#define CH   24      // CDIM / HEADS

typedef __attribute__((ext_vector_type(2))) float v2f;
typedef __attribute__((ext_vector_type(8))) float v8f;

__device__ __forceinline__ v8f wmma_f32_4(v2f a, v2f b, v8f c) {
  // (neg_a, A, neg_b, B, c_mod, C, reuse_a, reuse_b)
  return __builtin_amdgcn_wmma_f32_16x16x4_f32(false, a, false, b, (short)0, c,
                                               false, false);
}

// One DWORD global->LDS async copy, GVS addressing (SGPR64 base + VGPR32
// offset). LDS destination is the per-wave ds offset = low 32 bits of the
// generic shared-memory address (ISA 10.2 aperture mapping). Tracked on
// ASYNCcnt.
__device__ __forceinline__ void async_ld_b32(uint32_t lds_addr,
                                             const float* sbase,
                                             uint32_t voff_bytes) {
  asm volatile("global_load_async_to_lds_b32 %0, %1, %2"
               :: "v"(lds_addr), "v"(voff_bytes), "s"(sbase) : "memory");
}
__device__ __forceinline__ void wait_async0() {
  asm volatile("s_wait_asynccnt 0" ::: "memory");
}

// ---------------------------------------------------------------------------
// Per-pixel LayerNorm stats over the 192 channels: stats[p] = (mu, rstd).
// One thread per pixel; coalesced across every strided channel row.
// ---------------------------------------------------------------------------
__global__ __launch_bounds__(256) void ln_stats_kernel(
    const float* __restrict__ src, float2* __restrict__ stats) {
  int p = blockIdx.x * 256 + threadIdx.x;       // 0 .. 2*HW-1
  int b = p / HW, i = p % HW;
  const float* s = src + (long)b * CDIM * HW + i;
  float sum = 0.f, sumsq = 0.f;
  for (int c = 0; c < CDIM; ++c) {
    float v = s[(long)c * HW];
    sum += v; sumsq += v * v;
  }
  float mu  = sum * (1.0f / CDIM);
  float var = sumsq * (1.0f / CDIM) - mu * mu;
  stats[p] = make_float2(mu, rsqrtf(var + 1e-5f));
}

// ---------------------------------------------------------------------------
// Generic GEMM: out[z][m][n] = sum_k W[z][m][k] * X[z][k][n] + bias[m]
// N == HW. Block: 256 thr = 8 waves; tile 64(M) x 32(N); K staged in LDS x32.
// A tile: async global->LDS copies. B tile: VGPR staging with optional fused
// LayerNorm ((x-mu)*rstd*lnw[k]+lnb[k]) using precomputed per-pixel stats.
// ---------------------------------------------------------------------------
__global__ __launch_bounds__(256) void gemm_bias_kernel(
    const float* __restrict__ W, long wBS,
    const float* __restrict__ X, long xBS,
    const float2* __restrict__ lnstats,
    const float* __restrict__ lnw, const float* __restrict__ lnb,
    const float* __restrict__ bias, float* __restrict__ out, long oBS,
    int M, int K) {
  const int N = HW;
  __shared__ float As[64][34];   // padded rows -> conflict-free
  __shared__ float Bs[32][34];
  int tid  = threadIdx.x;
  int wave = tid >> 5, lane = tid & 31;
  int hi = lane >> 4, ln = lane & 15;     // hi selects the K pair (ISA layout)
  int wm = wave >> 1, wn = wave & 1;      // 4x2 wave grid
  int m0 = blockIdx.y * 64;
  int n0 = blockIdx.x * 32;
  const float* Wb = W + (long)blockIdx.z * wBS;
  const float* Xb = X + (long)blockIdx.z * xBS;
  float* Ob = out + (long)blockIdx.z * oBS;

  v8f acc = {};
  for (int kk = 0; kk < K; kk += 32) {
    // A: 64x32 weight tile via async copies (memory -> LDS, no VGPR staging)
#pragma unroll
    for (int it = 0; it < 8; ++it) {
      int r = wave + it * 8;              // 0..63
      uint32_t lds_a = (uint32_t)(uintptr_t)&As[r][lane];
      uint32_t go = (uint32_t)(((long)(m0 + r) * K + kk + lane) * 4);
      async_ld_b32(lds_a, Wb, go);
    }
    // B: 32x32 activation tile, optional fused LayerNorm
#pragma unroll
    for (int it = 0; it < 4; ++it) {
      int r = wave + it * 8;              // 0..31 (k index)
      float v = Xb[(long)(kk + r) * N + n0 + lane];
      if (lnstats) {
        float2 st = lnstats[(long)blockIdx.z * HW + n0 + lane];
        v = (v - st.x) * st.y * lnw[kk + r] + lnb[kk + r];
      }
      Bs[r][lane] = v;
    }
    wait_async0();
    __syncthreads();
#pragma unroll
    for (int k4 = 0; k4 < 32; k4 += 4) {
      int ka = k4 + 2 * hi;
      v2f a, bb;
      a.x  = As[16 * wm + ln][ka];
      a.y  = As[16 * wm + ln][ka + 1];
      bb.x = Bs[ka][16 * wn + ln];
      bb.y = Bs[ka + 1][16 * wn + ln];
      acc = wmma_f32_4(a, bb, acc);
    }
    __syncthreads();
  }
#pragma unroll
  for (int r = 0; r < 8; ++r) {
    int mrow = m0 + 16 * wm + r + 8 * hi;
    int ncol = n0 + 16 * wn + ln;
    Ob[(long)mrow * N + ncol] = acc[r] + bias[mrow];
  }
}

// ---------------------------------------------------------------------------
// Depthwise 3x3 conv, pad 1, virtual concat (c < split from srcA, rest from
// srcB) and optional exact GELU.
// ---------------------------------------------------------------------------
__global__ __launch_bounds__(256) void dw3x3_kernel(
    const float* __restrict__ srcA, long bsA,
    const float* __restrict__ srcB, long bsB, int split,
    const float* __restrict__ wdw, const float* __restrict__ bias,
    float* __restrict__ out, int C, int gelu) {
  long idx = (long)blockIdx.x * 256 + threadIdx.x;
  int p = (int)(idx % HW);
  int c = (int)((idx / HW) % C);
  int b = (int)(idx / ((long)HW * C));
  int yy = p >> 7, xx = p & 127;
  const float* s = (c < split) ? (srcA + b * bsA + (long)c * HW)
                               : (srcB + b * bsB + (long)(c - split) * HW);
  const float* w9 = wdw + c * 9;
  float acc = bias[c];
#pragma unroll
  for (int dy = -1; dy <= 1; ++dy) {
    int y2 = yy + dy;
    if (y2 < 0 || y2 > 127) continue;
#pragma unroll
    for (int dx = -1; dx <= 1; ++dx) {
      int x2 = xx + dx;
      if (x2 < 0 || x2 > 127) continue;
      acc += w9[(dy + 1) * 3 + (dx + 1)] * s[y2 * WPX + x2];
    }
  }
  if (gelu) acc = 0.5f * acc * (1.0f + erff(acc * 0.70710678118654752f));
  out[(long)b * C * HW + (long)c * HW + p] = acc;
}

// ---------------------------------------------------------------------------
// Per-row 1/max(||row||, 1e-12). Blocks 0..383: q rows (qc);
// blocks 384..767: k rows (channels 192..383 of qkv2).
// ---------------------------------------------------------------------------
__global__ __launch_bounds__(256) void rowinvnorm_kernel(
    const float* __restrict__ qc, const float* __restrict__ qkv2,
    float* __restrict__ qn, float* __restrict__ kn) {
  int r = blockIdx.x;
  const float* src; float* dst;
  if (r < 384) {
    int b = r / CDIM, c = r % CDIM;
    src = qc + (long)b * CDIM * HW + (long)c * HW;
    dst = qn + r;
  } else {
    int rk = r - 384;
    int b = rk / CDIM, c = rk % CDIM;
    src = qkv2 + (long)b * 576 * HW + (long)(CDIM + c) * HW;
    dst = kn + rk;
  }
  float s = 0.f;
  for (int n = threadIdx.x; n < HW; n += 256) { float v = src[n]; s += v * v; }
  for (int o = 16; o > 0; o >>= 1) s += __shfl_xor(s, o, 32);
  __shared__ float ws[8];
  if ((threadIdx.x & 31) == 0) ws[threadIdx.x >> 5] = s;
  __syncthreads();
  if (threadIdx.x == 0) {
    float t = 0.f;
    for (int i = 0; i < 8; ++i) t += ws[i];
    *dst = 1.0f / fmaxf(sqrtf(t), 1e-12f);
  }
}

// ---------------------------------------------------------------------------
// Gram: S[bh][i][j] += sum_n q[i][n]*k[j][n], 24x24 padded to 32x32.
// 4 waves per block (2x2 quadrants), hw split 8-way with float atomics.
// ---------------------------------------------------------------------------
__global__ __launch_bounds__(128) void gram_kernel(
    const float* __restrict__ qc, const float* __restrict__ qkv2,
    float* __restrict__ S) {
  __shared__ float Qs[32][34];
  __shared__ float Ks[32][34];
  int tid  = threadIdx.x;
  int wave = tid >> 5, lane = tid & 31;
  int hi = lane >> 4, ln = lane & 15;
  int wi = wave >> 1, wj = wave & 1;
  int bh = blockIdx.y, b = bh >> 3, h = bh & 7;
  const float* qb = qc   + (long)b * CDIM * HW + (long)h * CH * HW;
  const float* kb = qkv2 + (long)b * 576 * HW + (long)(CDIM + h * CH) * HW;
  int nbase = blockIdx.x * 2048;

  v8f acc = {};
  for (int nt = 0; nt < 2048; nt += 32) {
    int n0 = nbase + nt;
#pragma unroll
    for (int it = 0; it < 8; ++it) {
      int l = tid + it * 128;
      int r = l >> 5, cc = l & 31;
      Qs[r][cc] = (r < CH) ? qb[(long)r * HW + n0 + cc] : 0.0f;
      Ks[r][cc] = (r < CH) ? kb[(long)r * HW + n0 + cc] : 0.0f;
    }
    __syncthreads();
#pragma unroll
    for (int k4 = 0; k4 < 32; k4 += 4) {
      int ka = k4 + 2 * hi;
      v2f a, bb;
      a.x  = Qs[16 * wi + ln][ka];
      a.y  = Qs[16 * wi + ln][ka + 1];
      bb.x = Ks[16 * wj + ln][ka];     // B[n][j] = Ks[j][n]
      bb.y = Ks[16 * wj + ln][ka + 1];
      acc = wmma_f32_4(a, bb, acc);
    }
    __syncthreads();
  }
#pragma unroll
  for (int r = 0; r < 8; ++r) {
    int i = 16 * wi + r + 8 * hi;
    int j = 16 * wj + ln;
    if (i < CH && j < CH)
      atomicAdd(&S[(long)bh * 1024 + i * 32 + j], acc[r]);
  }
}

// ---------------------------------------------------------------------------
// Softmax over 24-wide rows; folds 1/(||q_i|| ||k_j||) * temperature[h].
// One wave per (bh, i) row.
// ---------------------------------------------------------------------------
__global__ __launch_bounds__(32) void softmax_kernel(
    const float* __restrict__ S, const float* __restrict__ qn,
    const float* __restrict__ kn, const float* __restrict__ temp,
    float* __restrict__ P) {
  int r = blockIdx.x;               // 0..383
  int bh = r / CH, i = r % CH;
  int b = bh >> 3, h = bh & 7;
  int j = threadIdx.x;
  float v = -3.4e38f;
  if (j < CH)
    v = S[(long)bh * 1024 + i * 32 + j] *
        qn[b * CDIM + h * CH + i] * kn[b * CDIM + h * CH + j] * temp[h];
  float mx = v;
  for (int o = 16; o > 0; o >>= 1) mx = fmaxf(mx, __shfl_xor(mx, o, 32));
  float e = (j < CH) ? expf(v - mx) : 0.0f;
  float sum = e;
  for (int o = 16; o > 0; o >>= 1) sum += __shfl_xor(sum, o, 32);
  if (j < CH) P[(long)bh * 576 + i * CH + j] = e / sum;
}

// ---------------------------------------------------------------------------
// E[b][m][h*24+j] = sum_i Wproj[m][h*24+i] * P[b,h][i][j]
// (Wproj composed with the block-diagonal attention matrix, so the attention
// apply + projection collapse into a single GEMM over v.)
// ---------------------------------------------------------------------------
__global__ __launch_bounds__(256) void build_E_kernel(
    const float* __restrict__ projw, const float* __restrict__ P,
    float* __restrict__ E) {
  int idx = blockIdx.x * 256 + threadIdx.x;   // 0 .. 2*192*192-1
  int cj = idx % CDIM;
  int m  = (idx / CDIM) % CDIM;
  int b  = idx / (CDIM * CDIM);
  int h = cj / CH, j = cj % CH;
  const float* wrow = projw + (long)m * CDIM + h * CH;
  const float* prow = P + (long)(b * 8 + h) * 576 + j;
  float acc = 0.f;
#pragma unroll
  for (int i = 0; i < CH; ++i) acc += wrow[i] * prow[i * CH];
  E[idx] = acc;
}

// ---------------------------------------------------------------------------
extern "C" void kernel_launch(void* const* d_in, const int* in_sizes, int n_in,
                              void* d_out, int out_size, void* d_ws, size_t ws_size,
                              hipStream_t stream) {
  (void)in_sizes; (void)n_in; (void)out_size; (void)ws_size;
  const float* x        = (const float*)d_in[0];
  const float* y        = (const float*)d_in[1];
  const float* ln_w     = (const float*)d_in[2];
  const float* ln_b     = (const float*)d_in[3];
  const float* qkv_w    = (const float*)d_in[4];
  const float* qkv_b    = (const float*)d_in[5];
  const float* qkv_dw_w = (const float*)d_in[6];
  const float* qkv_dw_b = (const float*)d_in[7];
  const float* convy_w  = (const float*)d_in[8];
  const float* convy_b  = (const float*)d_in[9];
  const float* qdw1_w   = (const float*)d_in[10];
  const float* qdw1_b   = (const float*)d_in[11];
  const float* qdw2_w   = (const float*)d_in[12];
  const float* qdw2_b   = (const float*)d_in[13];
  const float* proj_w   = (const float*)d_in[14];
  const float* proj_b   = (const float*)d_in[15];
  const float* temp     = (const float*)d_in[16];
  float* out = (float*)d_out;

  const long HW_ = HW;
  float* ws     = (float*)d_ws;
  float*  qkv1g = ws;                         // qkv1 [2][576][HW], later g [2][384][HW]
  float*  qkv2  = qkv1g + 2L * 576 * HW_;     // [2][576][HW] (q|k|v)
  float*  yq    = qkv2  + 2L * 576 * HW_;     // y_q [2][192][HW]
  float*  qc    = yq    + 2L * 192 * HW_;     // qc  [2][192][HW]
  float2* statsx = (float2*)(qc + 2L * 192 * HW_);   // [2*HW] (mu, rstd)
  float2* statsy = statsx + 2L * HW_;
  float*  Sbuf  = (float*)(statsy + 2L * HW_);       // [16][32][32]
  float*  Pbuf  = Sbuf + 16L * 1024;                 // [16][24][24]
  float*  qn    = Pbuf + 16L * 576;                  // [384]
  float*  kn    = qn + 384;                          // [384]
  float*  Ebuf  = kn + 384;                          // [2][192][192]

  dim3 b256(256);
  // 1. LayerNorm stats (LN itself is fused into the GEMM B loads)
  ln_stats_kernel<<<128, b256, 0, stream>>>(x, statsx);
  ln_stats_kernel<<<128, b256, 0, stream>>>(y, statsy);
  // 2. qkv1 = Wqkv(576x192) @ LN(x)
  gemm_bias_kernel<<<dim3(512, 9, 2), b256, 0, stream>>>(
      qkv_w, 0, x, 192 * HW_, statsx, ln_w, ln_b, qkv_b, qkv1g, 576 * HW_,
      576, 192);
  // 3. qkv2 = dw3x3(qkv1), groups=576
  dw3x3_kernel<<<(unsigned)(2L * 576 * HW_ / 256), b256, 0, stream>>>(
      qkv1g, 576 * HW_, qkv1g, 576 * HW_, 576, qkv_dw_w, qkv_dw_b, qkv2, 576, 0);
  // 4. y_q = Wy(192x192) @ LN(y)
  gemm_bias_kernel<<<dim3(512, 3, 2), b256, 0, stream>>>(
      convy_w, 0, y, 192 * HW_, statsy, ln_w, ln_b, convy_b, yq, 192 * HW_,
      192, 192);
  // 5. g = gelu(dw3x3(concat(q, y_q))), groups=384  (into the dead qkv1 region)
  dw3x3_kernel<<<(unsigned)(2L * 384 * HW_ / 256), b256, 0, stream>>>(
      qkv2, 576 * HW_, yq, 192 * HW_, 192, qdw1_w, qdw1_b, qkv1g, 384, 1);
  // 6. qc = Wq2(192x384) @ g
  gemm_bias_kernel<<<dim3(512, 3, 2), b256, 0, stream>>>(
      qdw2_w, 0, qkv1g, 384 * HW_, nullptr, nullptr, nullptr, qdw2_b, qc,
      192 * HW_, 192, 384);
  // 7. inverse L2 norms for q rows (qc) and k rows (qkv2[192:384])
  rowinvnorm_kernel<<<768, b256, 0, stream>>>(qc, qkv2, qn, kn);
  // 8. S = q @ k^T (unnormalized), WMMA over hw, 8-way split + atomics
  hipMemsetAsync(Sbuf, 0, 16 * 1024 * sizeof(float), stream);
  gram_kernel<<<dim3(8, 16), dim3(128), 0, stream>>>(qc, qkv2, Sbuf);
  // 9. P = softmax(S * invnorms * temperature)
  softmax_kernel<<<384, dim3(32), 0, stream>>>(Sbuf, qn, kn, temp, Pbuf);
  // 10. E = Wproj @ blockdiag(P)   (attention apply fused into projection)
  build_E_kernel<<<288, b256, 0, stream>>>(proj_w, Pbuf, Ebuf);
  // 11. out = E_b @ v + proj_b
  gemm_bias_kernel<<<dim3(512, 3, 2), b256, 0, stream>>>(
      Ebuf, (long)CDIM * CDIM, qkv2 + 384L * HW_, 576 * HW_,
      nullptr, nullptr, nullptr, proj_b, out, 192 * HW_, 192, 192);
}